// StratifiedRecurrence_91079076479617
// MI455X (gfx1250) — compile-verified
//
#include <hip/hip_runtime.h>
#include <math.h>

// Problem constants (from reference)
#define CC 8
#define BB 4
#define TT 8192
#define DD 64
#define CHUNK 256          // timesteps per block
#define NCH (TT / CHUNK)   // 32 chunks
#define SEG 64             // timesteps per scan segment
#define NSEG (CHUNK / SEG) // 4 segments
#define LSTR 68            // padded LDS row stride (floats): bank-conflict-free
#define CBTD (CC * BB * TT * DD)

typedef __attribute__((ext_vector_type(2))) float v2f;
typedef __attribute__((ext_vector_type(8))) float v8f;

__device__ __forceinline__ float lambda_of(int c) {
  // matches python: 1.0 - 2.0 ** (-(c * log2(8192)/7)), computed in double
  return (float)(1.0 - exp2(-(double)c * (13.0 / 7.0)));
}

__device__ __forceinline__ float pow64f(float x) {
  float y = x * x;  // ^2
  y *= y;           // ^4
  y *= y;           // ^8
  y *= y;           // ^16
  y *= y;           // ^32
  y *= y;           // ^64
  return y;
}

// ---------------------------------------------------------------------------
// Kernel 1: gating matmul (WMMA f32 16x16x4) + sigmoid*x + chunk-local scan.
// grid = (NCH, B, 2*C) ; block = 256 (8 wave32)
// blockIdx.z: c = z & 7, dir = z >> 3  (dir 0 = fwd, 1 = bwd/reversed-time)
// ---------------------------------------------------------------------------
__global__ __launch_bounds__(256) void k_gate_localscan(
    const float* __restrict__ channels,
    const float* __restrict__ gwf, const float* __restrict__ gbf,
    const float* __restrict__ gwb, const float* __restrict__ gbb,
    float* __restrict__ out, float* __restrict__ wsS) {
  __shared__ float Wl[DD * LSTR];       // 64x64 weights, padded
  __shared__ float Xl[CHUNK * LSTR];    // 256x64 chunk (X, then gated, then scanned)
  __shared__ float biasS[DD];
  __shared__ float segS[NSEG * DD];     // per-segment local totals

  const int tid = threadIdx.x;
  const int wave = tid >> 5;
  const int lane = tid & 31;
  const int z = blockIdx.z;
  const int c = z & 7;
  const int dir = z >> 3;
  const int b = blockIdx.y;
  const int chunk = blockIdx.x;
  const int u0 = chunk * CHUNK;  // logical (direction-adjusted) time base

  const float* W = (dir ? gwb : gwf) + c * DD * DD;
  const float* gb = (dir ? gbb : gbf) + c * DD;
  const float* X = channels + (c * BB + b) * (TT * DD);

  // ---- cooperative loads ----
  for (int i = tid; i < DD * DD; i += 256)
    Wl[(i >> 6) * LSTR + (i & 63)] = W[i];
  if (tid < DD) biasS[tid] = gb[tid];
  for (int i = tid; i < CHUNK * DD; i += 256) {
    int row = i >> 6, col = i & 63;
    int u = u0 + row;
    int t = dir ? (TT - 1 - u) : u;
    Xl[row * LSTR + col] = X[t * DD + col];
  }
  __syncthreads();

  // ---- stage A: per-wave 16x64 gate tiles via V_WMMA_F32_16X16X4_F32 ----
  const int lh = lane >> 4;  // lane half selects K-pair / M+8
  const int ll = lane & 15;
  for (int tile = wave; tile < CHUNK / 16; tile += 8) {
    const int tb = tile * 16 * LSTR;
    // A fragments: A[m][k], lane = (k>>1)*16 + m, reg = k&1  -> float2 loads
    v2f a[16];
    const int aoff = tb + ll * LSTR + 2 * lh;
#pragma unroll
    for (int kk = 0; kk < 16; ++kk)
      a[kk] = *(const v2f*)&Xl[aoff + 4 * kk];

    v8f acc[4];
#pragma unroll
    for (int n = 0; n < 4; ++n) acc[n] = (v8f)0.0f;

#pragma unroll
    for (int n = 0; n < 4; ++n) {
      const int boff = (n * 16 + ll) * LSTR + 2 * lh;  // B[k][n] = W[e][d]
#pragma unroll
      for (int kk = 0; kk < 16; ++kk) {
        v2f bf = *(const v2f*)&Wl[boff + 4 * kk];
        acc[n] = __builtin_amdgcn_wmma_f32_16x16x4_f32(
            false, a[kk], false, bf, (short)0, acc[n], false, false);
      }
    }

    // epilogue: y = sigmoid(y + bias) * x, stored back in place
    // C/D layout: vgpr v, lanes0-15: M=v ; lanes16-31: M=v+8 ; N = lane&15
#pragma unroll
    for (int n = 0; n < 4; ++n) {
      const int e = n * 16 + ll;
      const float bv = biasS[e];
#pragma unroll
      for (int v = 0; v < 8; ++v) {
        const int row = v + 8 * lh;
        float y = acc[n][v] + bv;
        float g = 1.0f / (1.0f + expf(-y));
        float xv = Xl[tb + row * LSTR + e];
        Xl[tb + row * LSTR + e] = g * xv;
      }
    }
  }
  __syncthreads();

  // ---- stage B: chunk-local scan, 4 segments of 64 steps, per-d lanes ----
  const int d = tid & 63;
  const int seg = tid >> 6;
  const float lam = lambda_of(c);

  // pass 1: segment-local inclusive scan in LDS
  float h = 0.0f;
  const int r0 = seg * SEG;
#pragma unroll 4
  for (int k = 0; k < SEG; ++k) {
    float* p = &Xl[(r0 + k) * LSTR + d];
    h = lam * h + *p;
    *p = h;
  }
  segS[seg * DD + d] = h;
  __syncthreads();

  // carry into this segment: E_{seg-1} = sum_{j<seg} lam64^{seg-1-j} * S_j
  const float lam64 = pow64f(lam);
  float Cc = 0.0f;
  for (int j = 0; j < seg; ++j) Cc = lam64 * Cc + segS[j * DD + d];

  // chunk-local total (input-carry = 0) for cross-chunk scan
  if (seg == NSEG - 1) {
    float tot = lam64 * Cc + segS[(NSEG - 1) * DD + d];
    wsS[((z * BB + b) * NCH + chunk) * DD + d] = tot;
  }

  // pass 2: write chunk-local result + intra-chunk segment carry
  const int obase = (dir ? CBTD : 0) + (c * BB + b) * (TT * DD);
  float lp = lam;  // lam^{k+1}
#pragma unroll 4
  for (int k = 0; k < SEG; ++k) {
    int u = u0 + r0 + k;
    int t = dir ? (TT - 1 - u) : u;
    float val = Xl[(r0 + k) * LSTR + d] + lp * Cc;
    out[obase + t * DD + d] = val;
    lp *= lam;
  }
}

// ---------------------------------------------------------------------------
// Kernel 2: scan chunk carries over NCH chunks per (dir,c,b,d) sequence.
// 2*C*B*D = 4096 sequences; exclusive prefix stored to wsP.
// ---------------------------------------------------------------------------
__global__ __launch_bounds__(256) void k_chunk_scan(
    const float* __restrict__ wsS, float* __restrict__ wsP) {
  const int gid = blockIdx.x * 256 + threadIdx.x;  // 0..4095
  const int z = gid >> 8;
  const int b = (gid >> 6) & 3;
  const int d = gid & 63;
  const int c = z & 7;

  const float lam = lambda_of(c);
  float lamL = pow64f(lam);  // lam^64
  lamL = lamL * lamL;        // ^128
  lamL = lamL * lamL;        // ^256 = lam^CHUNK

  const int base = ((z * BB + b) * NCH) * DD + d;
  float E = 0.0f;
#pragma unroll 4
  for (int ch = 0; ch < NCH; ++ch) {
    const int i = base + ch * DD;
    wsP[i] = E;               // exclusive: global h at end of previous chunk
    E = lamL * E + wsS[i];
  }
}

// ---------------------------------------------------------------------------
// Kernel 3: apply cross-chunk carry: out += lam^{pos+1} * P_chunk
// Same grid mapping as kernel 1 stage B.
// ---------------------------------------------------------------------------
__global__ __launch_bounds__(256) void k_apply_carry(
    const float* __restrict__ wsP, float* __restrict__ out) {
  const int z = blockIdx.z;
  const int c = z & 7;
  const int dir = z >> 3;
  const int b = blockIdx.y;
  const int chunk = blockIdx.x;
  if (chunk == 0) return;          // no carry into first chunk
  const float lam = lambda_of(c);  // c==0 -> lam==0 -> nothing to add
  if (c == 0) return;

  const int tid = threadIdx.x;
  const int d = tid & 63;
  const int seg = tid >> 6;
  const float P = wsP[((z * BB + b) * NCH + chunk) * DD + d];

  const float lam64 = pow64f(lam);
  float lp = lam;                       // lam^{seg*64 + k + 1}, k = 0
  for (int j = 0; j < seg; ++j) lp *= lam64;

  const int u0 = chunk * CHUNK;
  const int obase = (dir ? CBTD : 0) + (c * BB + b) * (TT * DD);
#pragma unroll 4
  for (int k = 0; k < SEG; ++k) {
    int u = u0 + seg * SEG + k;
    int t = dir ? (TT - 1 - u) : u;
    out[obase + t * DD + d] += lp * P;
    lp *= lam;
  }
}

// ---------------------------------------------------------------------------
extern "C" void kernel_launch(void* const* d_in, const int* in_sizes, int n_in,
                              void* d_out, int out_size, void* d_ws, size_t ws_size,
                              hipStream_t stream) {
  const float* channels = (const float*)d_in[0];
  const float* gwf = (const float*)d_in[1];
  const float* gbf = (const float*)d_in[2];
  const float* gwb = (const float*)d_in[3];
  const float* gbb = (const float*)d_in[4];
  float* out = (float*)d_out;

  // workspace: chunk totals + exclusive prefixes, 2 * (2*C*B*NCH*D) floats = 1 MB
  float* wsS = (float*)d_ws;
  float* wsP = wsS + 2 * CC * BB * NCH * DD;

  dim3 grid(NCH, BB, 2 * CC);
  k_gate_localscan<<<grid, 256, 0, stream>>>(channels, gwf, gbf, gwb, gbb, out, wsS);
  k_chunk_scan<<<16, 256, 0, stream>>>(wsS, wsP);
  k_apply_carry<<<grid, 256, 0, stream>>>(wsP, out);
}